// ManifoldMatchingLoss_13194139533849
// MI455X (gfx1250) — compile-verified
//
#include <hip/hip_runtime.h>
#include <hip/hip_bf16.h>
#include <math.h>

typedef __attribute__((ext_vector_type(2))) float v2f;
typedef __attribute__((ext_vector_type(8))) float v8f;

#define BB 128   // batch size (rows of the adjacency matrices)

// ---------------------------------------------------------------------------
// Kernel 1: G = F * F^T for a (BB x K) fp32 matrix, via V_WMMA_F32_16X16X4_F32.
// One 16x16 output tile per block; 4 wave32s split the K reduction.
// 4 independent accumulators per wave break the WMMA->WMMA RAW chain so the
// XDL pipe can issue back-to-back instead of stalling on accumulate latency.
// ---------------------------------------------------------------------------
__global__ void gram_wmma(const float* __restrict__ F, int K, float* __restrict__ G) {
    const int bn   = blockIdx.x;          // column tile (16 cols)
    const int bm   = blockIdx.y;          // row tile    (16 rows)
    const int tid  = threadIdx.x;
    const int wave = tid >> 5;            // 0..3
    const int lane = tid & 31;
    const int mn   = lane & 15;           // M index for A, N index for B
    const int hi   = lane >> 4;           // selects K pair {0,1} vs {2,3}

    const int kseg = K >> 2;              // K split over 4 waves (2048 / 1024)
    const int k0   = wave * kseg;
    const int k1   = k0 + kseg;

    // A[m][k] layout (f32 16x4): v0 = A[m][kb+2*hi], v1 = A[m][kb+2*hi+1]
    // B[k][n] = F[col_base+n][k] -> identical per-lane addressing, other row base.
    const float* arow = F + (size_t)(bm * 16 + mn) * (size_t)K + 2 * hi;
    const float* brow = F + (size_t)(bn * 16 + mn) * (size_t)K + 2 * hi;

    v8f acc0 = {}, acc1 = {}, acc2 = {}, acc3 = {};
    // kseg is a multiple of 16 (K in {8192, 4096}), no remainder.
    for (int k = k0; k < k1; k += 16) {
        v2f a0 = *(const v2f*)(arow + k);
        v2f b0 = *(const v2f*)(brow + k);
        v2f a1 = *(const v2f*)(arow + k + 4);
        v2f b1 = *(const v2f*)(brow + k + 4);
        v2f a2 = *(const v2f*)(arow + k + 8);
        v2f b2 = *(const v2f*)(brow + k + 8);
        v2f a3 = *(const v2f*)(arow + k + 12);
        v2f b3 = *(const v2f*)(brow + k + 12);
        // 8 args: (neg_a, A, neg_b, B, c_mod, C, reuse_a, reuse_b)
        acc0 = __builtin_amdgcn_wmma_f32_16x16x4_f32(false, a0, false, b0, (short)0, acc0, false, false);
        acc1 = __builtin_amdgcn_wmma_f32_16x16x4_f32(false, a1, false, b1, (short)0, acc1, false, false);
        acc2 = __builtin_amdgcn_wmma_f32_16x16x4_f32(false, a2, false, b2, (short)0, acc2, false, false);
        acc3 = __builtin_amdgcn_wmma_f32_16x16x4_f32(false, a3, false, b3, (short)0, acc3, false, false);
    }

    // Fold the 4 accumulators and reduce the 4 per-wave partials through LDS.
    __shared__ float red[4][32][8];
#pragma unroll
    for (int v = 0; v < 8; ++v)
        red[wave][lane][v] = (acc0[v] + acc1[v]) + (acc2[v] + acc3[v]);
    __syncthreads();

    if (wave == 0) {
#pragma unroll
        for (int v = 0; v < 8; ++v) {
            float s = (red[0][lane][v] + red[1][lane][v]) +
                      (red[2][lane][v] + red[3][lane][v]);
            // C/D layout: row M = v + 8*hi, col N = lane & 15
            int m = v + 8 * hi;
            G[(size_t)(bm * 16 + m) * BB + (bn * 16 + mn)] = s;
        }
    }
}

// ---------------------------------------------------------------------------
// Kernel 2: distances -> row log_softmax -> KL(column sums).
// pre[i][j] = -sqrt(max(sq,0)) * exp(T); pre[i][i] == 0 is the row max,
// so single-pass logsumexp is numerically stable.
// ---------------------------------------------------------------------------
__global__ void finalize_kl(const float* __restrict__ Gf, const float* __restrict__ Gg,
                            const float* __restrict__ tptr, float* __restrict__ out) {
    __shared__ float nf[BB], ng[BB], lf[BB], lg[BB];
    const int t = threadIdx.x;                 // 0..127
    const float s = expf(tptr[0]);             // exp(temperature)

    nf[t] = Gf[(size_t)t * BB + t];            // ||f_t||^2
    ng[t] = Gg[(size_t)t * BB + t];
    __syncthreads();

    // Phase 1: row i = t, logsumexp over j for both adjacency matrices.
    float sumf = 0.f, sumg = 0.f;
    for (int j = 0; j < BB; ++j) {
        float sqf = nf[t] + nf[j] - 2.f * Gf[(size_t)t * BB + j];
        float df  = (sqf > 0.f) ? sqrtf(sqf) : 0.f;
        sumf += expf(-df * s);
        float sqg = ng[t] + ng[j] - 2.f * Gg[(size_t)t * BB + j];
        float dg  = (sqg > 0.f) ? sqrtf(sqg) : 0.f;
        sumg += expf(-dg * s);
    }
    lf[t] = logf(sumf);                        // row-max is exactly 0
    lg[t] = logf(sumg);
    __syncthreads();

    // Phase 2: column j = t, accumulate KL over rows i (coalesced reads).
    float acc = 0.f;
    for (int i = 0; i < BB; ++i) {
        float sqf = nf[i] + nf[t] - 2.f * Gf[(size_t)i * BB + t];
        float df  = (sqf > 0.f) ? sqrtf(sqf) : 0.f;
        float Af  = -df * s - lf[i];
        float sqg = ng[i] + ng[t] - 2.f * Gg[(size_t)i * BB + t];
        float dg  = (sqg > 0.f) ? sqrtf(sqg) : 0.f;
        float Ag  = -dg * s - lg[i];
        acc += expf(Ag) * (Ag - Af);
    }
    out[t] = acc;
}

// ---------------------------------------------------------------------------
// Launch: inputs are image_feats (128*64*128 f32), lang_feats (128*32*128 f32),
// temperature (1 f32). Output: 128 f32. Workspace: two 128x128 Gram matrices.
// ---------------------------------------------------------------------------
extern "C" void kernel_launch(void* const* d_in, const int* in_sizes, int n_in,
                              void* d_out, int out_size, void* d_ws, size_t ws_size,
                              hipStream_t stream) {
    const float* img  = (const float*)d_in[0];
    const float* lng  = (const float*)d_in[1];
    const float* temp = (const float*)d_in[2];
    float* out = (float*)d_out;

    float* gf = (float*)d_ws;               // 128*128 f32 = 64 KB
    float* gg = gf + BB * BB;               // next 64 KB

    const int Kimg  = in_sizes[0] / BB;     // 8192
    const int Klang = in_sizes[1] / BB;     // 4096

    dim3 grid(BB / 16, BB / 16);            // 8x8 tiles of 16x16
    gram_wmma<<<grid, 128, 0, stream>>>(img, Kimg, gf);
    gram_wmma<<<grid, 128, 0, stream>>>(lng, Klang, gg);
    finalize_kl<<<1, BB, 0, stream>>>(gf, gg, temp, out);
}